// CrossModalFusion_77249281786180
// MI455X (gfx1250) — compile-verified
//
#include <hip/hip_runtime.h>
#include <math.h>

typedef __attribute__((ext_vector_type(2))) float v2f;
typedef __attribute__((ext_vector_type(8))) float v8f;

#define BB   8
#define CC   512
#define HWD  4096
#define NTOT (BB * CC * HWD)      // 16,777,216 fused elements
#define LOSS_BLOCKS 1024

// ---------------------------------------------------------------------------
// WMMA K-loop: accumulate a 16x16 (MxN) f32 tile of A(8xK) * W^T using
// V_WMMA_F32_16X16X4_F32. A rows 8..15 are *duplicates* of rows 0..7
// (l & 7) -> unconditional loads, no EXEC-masked branches; the duplicate
// output rows land in accumulator lanes 16..31 which are never stored.
// Fragment layouts per CDNA5 ISA 7.12.2:
//   A 16x4: lanes 0-15 rows M=0..15, K={k0,k0+1}; lanes 16-31 K={k0+2,k0+3}
//   B 4x16: N striped over lanes, K halves over lane groups (symmetric)
//   C/D:    VGPR r: lanes 0-15 -> (M=r, N=lane); lanes 16-31 -> (M=r+8)
// ---------------------------------------------------------------------------
__device__ __forceinline__ v8f wmma_kloop_f32(v8f acc,
    const float* __restrict__ A, int lda,
    const float* __restrict__ W, int ldw,
    int n0, int K, int lane)
{
  const int l    = lane & 15;
  const int half = lane >> 4;
  const float* arow = A + (l & 7) * lda;            // pad lanes read dup rows
  const float* wrow = W + (long)(n0 + l) * ldw;
#pragma unroll 4
  for (int k0 = 0; k0 < K; k0 += 4) {
    const int ka = k0 + 2 * half;
    v2f a, b;
    a.x = arow[ka];
    a.y = arow[ka + 1];
    b.x = wrow[ka];
    b.y = wrow[ka + 1];
    acc = __builtin_amdgcn_wmma_f32_16x16x4_f32(false, a, false, b,
                                                (short)0, acc, false, false);
  }
  return acc;
}

// Store the valid 8xN rows of the accumulator tile (+bias). Lanes 16-31 hold
// duplicate rows M=8..15 and are skipped (branch is after all WMMAs).
__device__ __forceinline__ void store_tile8(const v8f acc,
    float* __restrict__ out, int ldo,
    const float* __restrict__ bias, int n0, int lane)
{
  const int l    = lane & 15;
  const int half = lane >> 4;
  if (half == 0) {
    const int   n  = n0 + l;
    const float bv = bias[n];
#pragma unroll
    for (int r = 0; r < 8; ++r) out[r * ldo + n] = acc[r] + bv;
  }
}

// ---------------------------------------------------------------------------
// Stage kernels: one wave32 per 16-column tile so each stage's weight read
// (1-2 MB) is spread across many WGPs instead of one.
// ---------------------------------------------------------------------------
__global__ __launch_bounds__(32)
void gemm8_single(const float* __restrict__ A, const float* __restrict__ W,
                  const float* __restrict__ bias, float* __restrict__ out,
                  int K, int ldw)
{
  const int lane = threadIdx.x;
  const int n0   = blockIdx.x * 16;
  v8f acc = {};
  acc = wmma_kloop_f32(acc, A, K, W, ldw, n0, K, lane);
  store_tile8(acc, out, CC, bias, n0, lane);
}

// Two independent GEMMs (rgb / dep) in one launch: blocks [0,32) -> set 0,
// blocks [32,64) -> set 1. Uniform scalar select, no lane divergence.
__global__ __launch_bounds__(32)
void gemm8_dual(const float* __restrict__ A0, const float* __restrict__ W0,
                const float* __restrict__ b0, float* __restrict__ out0,
                const float* __restrict__ A1, const float* __restrict__ W1,
                const float* __restrict__ b1, float* __restrict__ out1)
{
  const int  lane   = threadIdx.x;
  const bool second = blockIdx.x >= 32;
  const int  n0     = (blockIdx.x & 31) * 16;
  const float* A    = second ? A1 : A0;
  const float* W    = second ? W1 : W0;
  const float* bias = second ? b1 : b0;
  float*       out  = second ? out1 : out0;
  v8f acc = {};
  acc = wmma_kloop_f32(acc, A, CC, W, CC, n0, CC, lane);
  store_tile8(acc, out, CC, bias, n0, lane);
}

// Gate pre-activation: K=1024 split across the concat halves [o_rgb | o_dep].
__global__ __launch_bounds__(32)
void gemm8_cat(const float* __restrict__ A0, const float* __restrict__ A1,
               const float* __restrict__ W /* 512 x 1024 */,
               const float* __restrict__ bias, float* __restrict__ out)
{
  const int lane = threadIdx.x;
  const int n0   = blockIdx.x * 16;
  v8f acc = {};
  acc = wmma_kloop_f32(acc, A0, CC, W,      2 * CC, n0, CC, lane);
  acc = wmma_kloop_f32(acc, A1, CC, W + CC, 2 * CC, n0, CC, lane);
  store_tile8(acc, out, CC, bias, n0, lane);
}

// ---------------------------------------------------------------------------
// BN-over-batch + sigmoid gate + fuse + spatial broadcast, fused in one pass.
// One block per (b,c); the 8 uniform `pre` loads for mean/var are L2-hot.
// Softmax over the single text key == 1, so rgb_t/dep_t are spatially
// constant and fused[b,c,:,:] is a single value broadcast over 64x64.
// ---------------------------------------------------------------------------
__global__ __launch_bounds__(256)
void gate_broadcast_kernel(const float* __restrict__ pre,
                           const float* __restrict__ o_rgb,
                           const float* __restrict__ o_dep,
                           const float* __restrict__ bn_g,
                           const float* __restrict__ bn_b,
                           float* __restrict__ out)
{
  const int bc = blockIdx.x;          // 0 .. B*C-1
  const int b  = bc >> 9;             // / 512
  const int c  = bc & (CC - 1);

  float mean = 0.0f, m2 = 0.0f;
#pragma unroll
  for (int m = 0; m < BB; ++m) {
    const float x = pre[m * CC + c];
    mean += x; m2 += x * x;
  }
  mean *= (1.0f / BB);
  const float var  = m2 * (1.0f / BB) - mean * mean;
  const float inv  = rsqrtf(var + 1e-5f);
  const float xhat = (pre[b * CC + c] - mean) * inv;
  const float gate = 1.0f / (1.0f + __expf(-(bn_g[c] * xhat + bn_b[c])));
  const float v    = gate * o_rgb[b * CC + c] + (1.0f - gate) * o_dep[b * CC + c];

  const float4 v4 = make_float4(v, v, v, v);
  float4* base = reinterpret_cast<float4*>(out + (size_t)bc * HWD);
#pragma unroll
  for (int i = 0; i < 4; ++i)
    base[threadIdx.x + i * 256] = v4;
}

// ---------------------------------------------------------------------------
// Deterministic two-stage reduction for pixel/depth losses (no float atomics)
// ---------------------------------------------------------------------------
__global__ __launch_bounds__(256)
void loss_partial_kernel(const float4* __restrict__ rgb,
                         const float4* __restrict__ dep,
                         float* __restrict__ partials)
{
  __shared__ float s_sq[256];
  __shared__ float s_ab[256];
  const int tid = threadIdx.x;
  float sq = 0.0f, ab = 0.0f;
  const int N4 = NTOT / 4;
  for (int i = blockIdx.x * 256 + tid; i < N4; i += LOSS_BLOCKS * 256) {
    const float4 r = rgb[i], d = dep[i];
    const float dx = r.x - d.x, dy = r.y - d.y;
    const float dz = r.z - d.z, dw = r.w - d.w;
    sq += dx * dx + dy * dy + dz * dz + dw * dw;
    ab += fabsf(dx) + fabsf(dy) + fabsf(dz) + fabsf(dw);
  }
  s_sq[tid] = sq; s_ab[tid] = ab;
  __syncthreads();
  for (int s = 128; s > 0; s >>= 1) {
    if (tid < s) { s_sq[tid] += s_sq[tid + s]; s_ab[tid] += s_ab[tid + s]; }
    __syncthreads();
  }
  if (tid == 0) {
    partials[blockIdx.x]               = s_sq[0];
    partials[LOSS_BLOCKS + blockIdx.x] = s_ab[0];
  }
}

__global__ __launch_bounds__(256)
void loss_final_kernel(const float* __restrict__ partials, float* __restrict__ out)
{
  __shared__ float s_sq[256];
  __shared__ float s_ab[256];
  const int tid = threadIdx.x;
  float sq = 0.0f, ab = 0.0f;
  for (int i = tid; i < LOSS_BLOCKS; i += 256) {
    sq += partials[i];
    ab += partials[LOSS_BLOCKS + i];
  }
  s_sq[tid] = sq; s_ab[tid] = ab;
  __syncthreads();
  for (int s = 128; s > 0; s >>= 1) {
    if (tid < s) { s_sq[tid] += s_sq[tid + s]; s_ab[tid] += s_ab[tid + s]; }
    __syncthreads();
  }
  if (tid == 0) {
    out[NTOT]     = s_sq[0] / (float)NTOT;   // pixel_loss
    out[NTOT + 1] = s_ab[0] / (float)NTOT;   // depth_loss
  }
}

// ---------------------------------------------------------------------------
extern "C" void kernel_launch(void* const* d_in, const int* in_sizes, int n_in,
                              void* d_out, int out_size, void* d_ws, size_t ws_size,
                              hipStream_t stream)
{
  const float* rgb_c4    = (const float*)d_in[0];
  const float* depth_c4  = (const float*)d_in[1];
  const float* text_feat = (const float*)d_in[2];
  const float* tp_w      = (const float*)d_in[3];
  const float* tp_b      = (const float*)d_in[4];
  const float* rgb_wqkv  = (const float*)d_in[5];
  const float* rgb_bqkv  = (const float*)d_in[6];
  const float* rgb_wo    = (const float*)d_in[7];
  const float* rgb_bo    = (const float*)d_in[8];
  const float* dep_wqkv  = (const float*)d_in[9];
  const float* dep_bqkv  = (const float*)d_in[10];
  const float* dep_wo    = (const float*)d_in[11];
  const float* dep_bo    = (const float*)d_in[12];
  const float* gate_w    = (const float*)d_in[13];
  const float* gate_b    = (const float*)d_in[14];
  const float* bn_g      = (const float*)d_in[15];
  const float* bn_b      = (const float*)d_in[16];

  float* out = (float*)d_out;
  float* ws  = (float*)d_ws;

  float* w_text    = ws;                 // 8x512 text_proj
  float* w_vr      = ws + 1 * BB * CC;   // v_rgb
  float* w_vd      = ws + 2 * BB * CC;   // v_dep
  float* w_or      = ws + 3 * BB * CC;   // o_rgb (== rgb_t, spatially const)
  float* w_od      = ws + 4 * BB * CC;   // o_dep
  float* w_pre     = ws + 5 * BB * CC;   // gate pre-activation
  float* partials  = ws + 6 * BB * CC;   // 2 * LOSS_BLOCKS

  // Stage 1: text_proj = text_feat @ tp_w^T + tp_b
  gemm8_single<<<32, 32, 0, stream>>>(text_feat, tp_w, tp_b, w_text, CC, CC);

  // Stage 2: V projections (Wv = wqkv rows [2C:3C], bv = bqkv[2C:])
  gemm8_dual<<<64, 32, 0, stream>>>(
      w_text, rgb_wqkv + 2 * CC * CC, rgb_bqkv + 2 * CC, w_vr,
      w_text, dep_wqkv + 2 * CC * CC, dep_bqkv + 2 * CC, w_vd);

  // Stage 3: output projections (attention over 1 key is identity on V)
  gemm8_dual<<<64, 32, 0, stream>>>(
      w_vr, rgb_wo, rgb_bo, w_or,
      w_vd, dep_wo, dep_bo, w_od);

  // Stage 4: pre = [o_rgb | o_dep] @ gate_w^T + gate_b
  gemm8_cat<<<32, 32, 0, stream>>>(w_or, w_od, gate_w, gate_b, w_pre);

  // Stage 5: BN + gate + fuse + 64x64 broadcast (64 MB write)
  gate_broadcast_kernel<<<BB * CC, 256, 0, stream>>>(
      w_pre, w_or, w_od, bn_g, bn_b, out);

  // Losses (128 MB read), deterministic two-stage tree
  loss_partial_kernel<<<LOSS_BLOCKS, 256, 0, stream>>>(
      (const float4*)rgb_c4, (const float4*)depth_c4, partials);
  loss_final_kernel<<<1, 256, 0, stream>>>(partials, out);
}